// StochasticTwoLayerRGCN_4733053960249
// MI455X (gfx1250) — compile-verified
//
#include <hip/hip_runtime.h>

// ---------------- problem constants (match reference) ----------------
#define RREL  4
#define INF   256
#define HIDF  128
#define OUTF  128
#define NSRC  200000
#define NMID  40000
#define NDST  8000

typedef float v2f __attribute__((ext_vector_type(2)));
typedef float v8f __attribute__((ext_vector_type(8)));

// =====================================================================
// WMMA f32 GEMM:  C[r] = A (MxK) * B[r] (KxN)   for r in [0, RREL)
// One 16x16 column tile per wave32, ALL 4 relations per wave: the A
// fragment (16x4) is loaded once per k-step and fed to 4 back-to-back
// v_wmma_f32_16x16x4_f32 ops (reuse-A hint on the middle two, where the
// previous instruction is the identical WMMA as §7.12 requires).
// A-fragment ISA layout: lanes 0-15 hold {K=k0,k0+1} for row M=lane,
// lanes 16-31 hold {K=k0+2,k0+3}; B-fragment (4x16) mirrored by half.
// C/D: VGPR v holds M=v (lanes 0-15, N=lane) and M=v+8 (lanes 16-31).
// M,N multiples of 16, K multiple of 4 at every call site => no
// divergence, EXEC all ones as WMMA requires.
// =====================================================================
__global__ void __launch_bounds__(128) gemm_wmma_f32(
    const float* __restrict__ A,   // [M][K]
    const float* __restrict__ B,   // [RREL][K][N]
    float* __restrict__ C,         // [RREL][M][N]
    int M, int K, int N)
{
  const int wave = threadIdx.x >> 5;            // 4 waves / block
  const int lane = threadIdx.x & 31;
  const int ct   = blockIdx.y * 4 + wave;       // 16-col tile index, 0..N/16-1
  const int row0 = blockIdx.x << 4;
  const int col0 = ct << 4;

  const int l16 = lane & 15;
  const int hi  = lane >> 4;                    // 0: K pair {0,1}, 1: {2,3}

  const float*  aptr  = A + (size_t)(row0 + l16) * K + (hi << 1);
  const size_t  rstep = (size_t)K * N;          // relation stride in B
  const float*  bptr  = B + (size_t)(hi << 1) * N + (col0 + l16);

  v8f acc0 = {}, acc1 = {}, acc2 = {}, acc3 = {};
  for (int k0 = 0; k0 < K; k0 += 4) {
    if ((k0 & 63) == 0) {
      __builtin_prefetch(aptr + 64, 0, 1);      // global_prefetch_b8, A stream
    }
    v2f a = *(const v2f*)aptr;                  // A[row][k0+2hi .. k0+2hi+1]
    v2f b0, b1, b2, b3;                         // B[r][k0+2hi(+1)][col]
    b0.x = bptr[0];             b0.y = bptr[N];
    b1.x = bptr[rstep];         b1.y = bptr[rstep + N];
    b2.x = bptr[2 * rstep];     b2.y = bptr[2 * rstep + N];
    b3.x = bptr[3 * rstep];     b3.y = bptr[3 * rstep + N];

    acc0 = __builtin_amdgcn_wmma_f32_16x16x4_f32(
        false, a, false, b0, (short)0, acc0, /*reuse_a=*/false, false);
    acc1 = __builtin_amdgcn_wmma_f32_16x16x4_f32(
        false, a, false, b1, (short)0, acc1, /*reuse_a=*/true,  false);
    acc2 = __builtin_amdgcn_wmma_f32_16x16x4_f32(
        false, a, false, b2, (short)0, acc2, /*reuse_a=*/true,  false);
    acc3 = __builtin_amdgcn_wmma_f32_16x16x4_f32(
        false, a, false, b3, (short)0, acc3, /*reuse_a=*/false, false);

    aptr += 4;
    bptr += (size_t)4 * N;
  }

  const size_t mn   = (size_t)M * N;
  const size_t coff = (size_t)(row0 + (hi << 3)) * N + (col0 + l16);
  float* c0 = C + coff;
  float* c1 = C + mn + coff;
  float* c2 = C + 2 * mn + coff;
  float* c3 = C + 3 * mn + coff;
#pragma unroll
  for (int v = 0; v < 8; ++v) {
    c0[(size_t)v * N] = acc0[v];
    c1[(size_t)v * N] = acc1[v];
    c2[(size_t)v * N] = acc2[v];
    c3[(size_t)v * N] = acc3[v];
  }
}

// =====================================================================
// Edge scatter: one thread per (relation, edge, feature f<128).
// Wave-coalesced 512B row gather from m; f32 atomic adds land in the
// L2-resident per-relation accumulator (82MB layer1 < 192MB L2).
// Lane f==0 fuses the degree count.
// =====================================================================
__global__ void __launch_bounds__(256) scatter_edges_f128(
    const float* __restrict__ m,    // [RREL][Nsrc][128]
    const int*   __restrict__ src,  // [RREL][E]
    const int*   __restrict__ dst,  // [RREL][E]
    float* __restrict__ agg,        // [RREL][Ndst][128]
    float* __restrict__ deg,        // [RREL][Ndst]
    int E, int Nsrc, int Ndst)
{
  size_t gid  = (size_t)blockIdx.x * 256 + threadIdx.x;
  size_t eAll = gid >> 7;                       // r*E + e
  if (eAll >= (size_t)RREL * (size_t)E) return;
  int f = (int)(gid & 127);
  int r = (int)(eAll / (size_t)E);
  int e = (int)(eAll - (size_t)r * (size_t)E);
  int s = src[(size_t)r * E + e];
  int d = dst[(size_t)r * E + e];
  float val = m[((size_t)r * Nsrc + s) * 128 + f];
  atomicAdd(&agg[((size_t)r * Ndst + d) * 128 + f], val);
  if (f == 0) atomicAdd(&deg[(size_t)r * Ndst + d], 1.0f);
}

// =====================================================================
// Epilogue: out[v][f] = sum_r agg[r][v][f] * (deg>0 ? 1/deg : 0) + b[r][f]
// (bias added unconditionally per relation, matching the reference)
// =====================================================================
__global__ void __launch_bounds__(256) finish_f128(
    const float* __restrict__ agg,  // [RREL][Ndst][128]
    const float* __restrict__ deg,  // [RREL][Ndst]
    const float* __restrict__ bias, // [RREL][128]
    float* __restrict__ out,        // [Ndst][128]
    int Ndst)
{
  size_t gid = (size_t)blockIdx.x * 256 + threadIdx.x;
  if (gid >= (size_t)Ndst * 128) return;
  int f = (int)(gid & 127);
  int v = (int)(gid >> 7);
  float accv = 0.f;
#pragma unroll
  for (int r = 0; r < RREL; ++r) {
    float dg  = deg[(size_t)r * Ndst + v];
    float nrm = dg > 0.f ? (1.0f / dg) : 0.f;
    accv += agg[((size_t)r * Ndst + v) * 128 + f] * nrm + bias[r * 128 + f];
  }
  out[gid] = accv;
}

// =====================================================================
// Launch: gemm1 -> scatter1 -> finish1(h) -> gemm2 -> scatter2 -> finish2
// Workspace layout (floats):
//   m1   [4][200000][128]   409.6 MB
//   agg1 [4][40000][128]  \
//   deg1 [4][40000]        |  contiguous 99.1 MB zero region (one memset)
//   agg2 [4][8000][128]    |
//   deg2 [4][8000]        /
//   h    [40000][128]       20.5 MB
//   m2   [4][40000][128]    81.9 MB
// Total ~611 MB of d_ws.
// =====================================================================
extern "C" void kernel_launch(void* const* d_in, const int* in_sizes, int n_in,
                              void* d_out, int out_size, void* d_ws, size_t ws_size,
                              hipStream_t stream)
{
  const float* x    = (const float*)d_in[0];
  const float* W1   = (const float*)d_in[1];
  const float* b1   = (const float*)d_in[2];
  const float* W2   = (const float*)d_in[3];
  const float* b2   = (const float*)d_in[4];
  const int*   src1 = (const int*)d_in[5];
  const int*   dst1 = (const int*)d_in[6];
  const int*   src2 = (const int*)d_in[7];
  const int*   dst2 = (const int*)d_in[8];
  const int E1 = in_sizes[5] / RREL;   // 1,000,000
  const int E2 = in_sizes[7] / RREL;   //   200,000

  float* ws = (float*)d_ws;
  size_t off = 0;
  float* m1   = ws + off; off += (size_t)RREL * NSRC * HIDF;
  float* agg1 = ws + off; off += (size_t)RREL * NMID * HIDF;
  float* deg1 = ws + off; off += (size_t)RREL * NMID;
  float* agg2 = ws + off; off += (size_t)RREL * NDST * OUTF;
  float* deg2 = ws + off; off += (size_t)RREL * NDST;
  float* h    = ws + off; off += (size_t)NMID * HIDF;
  float* m2   = ws + off; off += (size_t)RREL * NMID * OUTF;

  // zero agg1..deg2 (contiguous) every call; graph-capture-safe memset node
  size_t zeroBytes = ((size_t)RREL * NMID * HIDF + (size_t)RREL * NMID +
                      (size_t)RREL * NDST * OUTF + (size_t)RREL * NDST) * sizeof(float);
  hipMemsetAsync(agg1, 0, zeroBytes, stream);

  // ---- layer 1 ----
  {
    dim3 grid(NSRC / 16, (HIDF / 16) / 4);            // (12500, 2)
    gemm_wmma_f32<<<grid, 128, 0, stream>>>(x, W1, m1, NSRC, INF, HIDF);
  }
  {
    size_t total = (size_t)RREL * (size_t)E1 * 128;
    unsigned blocks = (unsigned)((total + 255) / 256);
    scatter_edges_f128<<<blocks, 256, 0, stream>>>(m1, src1, dst1, agg1, deg1,
                                                   E1, NSRC, NMID);
  }
  finish_f128<<<(NMID * 128) / 256, 256, 0, stream>>>(agg1, deg1, b1, h, NMID);

  // ---- layer 2 ----
  {
    dim3 grid(NMID / 16, (OUTF / 16) / 4);            // (2500, 2)
    gemm_wmma_f32<<<grid, 128, 0, stream>>>(h, W2, m2, NMID, HIDF, OUTF);
  }
  {
    size_t total = (size_t)RREL * (size_t)E2 * 128;
    unsigned blocks = (unsigned)((total + 255) / 256);
    scatter_edges_f128<<<blocks, 256, 0, stream>>>(m2, src2, dst2, agg2, deg2,
                                                   E2, NMID, NDST);
  }
  finish_f128<<<(NDST * 128) / 256, 256, 0, stream>>>(agg2, deg2, b2,
                                                      (float*)d_out, NDST);
}